// MMAKernelLayer_45483703665413
// MI455X (gfx1250) — compile-verified
//
#include <hip/hip_runtime.h>
#include <hip/hip_bf16.h>
#include <float.h>

typedef __attribute__((ext_vector_type(2))) float v2f;
typedef __attribute__((ext_vector_type(8))) float v8f;

#define RES 30
#define GSIZE (RES * RES)          // 900 grid points per sample
#define NI 32                      // intervals
#define NC 32                      // corners
#define META_STRIDE 36             // lox, loy, stepx, stepy, w2[32]

// ---------------------------------------------------------------------------
// Kernel 1: per-sample bbox -> (lo, step), and per-interval squared weights.
// One 256-thread block (8 waves) per sample.
// ---------------------------------------------------------------------------
__global__ __launch_bounds__(256) void pi_setup_kernel(
    const float* __restrict__ births, const float* __restrict__ deaths,
    float* __restrict__ meta) {
  const int s = blockIdx.x;
  const int t = threadIdx.x;
  __shared__ float smnx[256], smny[256], smxx[256], smxy[256];

  const float* bb = births + (size_t)s * (NI * NC * 2);
  const float* dd = deaths + (size_t)s * (NI * NC * 2);

  float mnx = FLT_MAX, mny = FLT_MAX, mxx = -FLT_MAX, mxy = -FLT_MAX;
  for (int p = t; p < NI * NC; p += 256) {
    float x = bb[2 * p], y = bb[2 * p + 1];
    mnx = fminf(mnx, x); mxx = fmaxf(mxx, x);
    mny = fminf(mny, y); mxy = fmaxf(mxy, y);
    x = dd[2 * p]; y = dd[2 * p + 1];
    mnx = fminf(mnx, x); mxx = fmaxf(mxx, x);
    mny = fminf(mny, y); mxy = fmaxf(mxy, y);
  }
  smnx[t] = mnx; smny[t] = mny; smxx[t] = mxx; smxy[t] = mxy;
  __syncthreads();
  for (int off = 128; off > 0; off >>= 1) {
    if (t < off) {
      smnx[t] = fminf(smnx[t], smnx[t + off]);
      smny[t] = fminf(smny[t], smny[t + off]);
      smxx[t] = fmaxf(smxx[t], smxx[t + off]);
      smxy[t] = fmaxf(smxy[t], smxy[t + off]);
    }
    __syncthreads();
  }

  float* m = meta + s * META_STRIDE;
  if (t == 0) {
    float rngx = smxx[0] - smnx[0];
    float rngy = smxy[0] - smny[0];
    // lo = bmin - 0.1*rng ; span = hi - lo = 1.2*rng ; step = span/(RES-1)
    m[0] = smnx[0] - 0.1f * rngx;
    m[1] = smny[0] - 0.1f * rngy;
    m[2] = (rngx * 1.2f) / (float)(RES - 1);
    m[3] = (rngy * 1.2f) / (float)(RES - 1);
  }
  // interval weights: w = mean_c max(|dx-bx|, |dy-by|); store w^2 (P = 2)
  if (t < NI) {
    const float* bi = bb + t * (NC * 2);
    const float* di = dd + t * (NC * 2);
    float acc = 0.0f;
    for (int c = 0; c < NC; ++c) {
      float ax = fabsf(di[2 * c]     - bi[2 * c]);
      float ay = fabsf(di[2 * c + 1] - bi[2 * c + 1]);
      acc += fmaxf(ax, ay);
    }
    float w = acc * (1.0f / (float)NC);
    m[4 + t] = w * w;
  }
}

// elementwise min of two 16x16 f32 accumulators, then in-lane 8->1 min tree
__device__ __forceinline__ float pairmin8(v8f a, v8f b) {
  float m0 = fminf(a[0], b[0]);
  float m1 = fminf(a[1], b[1]);
  float m2 = fminf(a[2], b[2]);
  float m3 = fminf(a[3], b[3]);
  float m4 = fminf(a[4], b[4]);
  float m5 = fminf(a[5], b[5]);
  float m6 = fminf(a[6], b[6]);
  float m7 = fminf(a[7], b[7]);
  m0 = fminf(m0, m4); m1 = fminf(m1, m5);
  m2 = fminf(m2, m6); m3 = fminf(m3, m7);
  m0 = fminf(m0, m2); m1 = fminf(m1, m3);
  return fminf(m0, m1);
}

// lane i <-> lane i^16 exchange + min, as pure VALU (v_permlanex16_b32 with
// identity select): avoids ds_bpermute + s_wait_dscnt in the hot loop.
__device__ __forceinline__ float xhalf_min(float x) {
  int xi = __float_as_int(x);
  int sw = __builtin_amdgcn_permlanex16(xi, xi, 0x76543210, 0xfedcba98,
                                        false, false);
  return fminf(x, __int_as_float(sw));
}

// ---------------------------------------------------------------------------
// Kernel 2: one wave32 per (sample, tile of 16 grid points).
// d^2 = |g|^2 + |p|^2 - 2 g.p realized as a K=4 f32 WMMA:
//   A row m  = [gx, gy, |g|^2, 1 ]        (16x4)
//   B col n  = [-2px, -2py, 1, |p|^2]     (4x16)
//   D[m][n]  = squared distance grid_m <-> corner_n
// 4 WMMAs per interval (birth/death x 2 corner tiles of 16).
// ---------------------------------------------------------------------------
__global__ __launch_bounds__(32) void pi_main_kernel(
    const float* __restrict__ births, const float* __restrict__ deaths,
    const float* __restrict__ meta, float* __restrict__ out) {
  const int tile = blockIdx.x;        // 57 tiles of 16 grid points
  const int s    = blockIdx.y;
  const int lane = threadIdx.x;       // wave32
  const int mrow = lane & 15;
  const bool hiHalf = lane >= 16;     // lanes 16..31 hold K=2,3 of A / B

  const float* mt = meta + s * META_STRIDE;
  const float lox = mt[0], loy = mt[1], sx = mt[2], sy = mt[3];

  const int g  = tile * 16 + mrow;
  const int gc = (g < GSIZE) ? g : (GSIZE - 1);
  const int ix = gc / RES;
  const int iy = gc - ix * RES;       // first axis (x) varies slowest
  const float gx = lox + (float)ix * sx;
  const float gy = loy + (float)iy * sy;
  const float r2 = gx * gx + gy * gy;

  // A (16x4 f32): lane m -> {K0,K1} = (gx,gy); lane m+16 -> {K2,K3} = (|g|^2,1)
  v2f A;
  A.x = hiHalf ? r2   : gx;
  A.y = hiHalf ? 1.0f : gy;

  const float2* pb = (const float2*)(births + (size_t)s * (NI * NC * 2));
  const float2* pd = (const float2*)(deaths + (size_t)s * (NI * NC * 2));

  float acc = 0.0f;
  for (int i = 0; i < NI; ++i) {
    const float2* bi = pb + i * NC;
    const float2* di = pd + i * NC;
    // corner tiles: cols 0..15 use corner (lane&15), cols 16..31 use +16
    float2 b0 = bi[mrow];
    float2 b1 = bi[mrow + 16];
    float2 d0 = di[mrow];
    float2 d1 = di[mrow + 16];

    // B (4x16 f32): lane n -> {K0,K1} = (-2px,-2py); lane n+16 -> {K2,K3} = (1,|p|^2)
    v2f Bb0, Bb1, Bd0, Bd1;
    Bb0.x = hiHalf ? 1.0f : -2.0f * b0.x;
    Bb0.y = hiHalf ? (b0.x * b0.x + b0.y * b0.y) : -2.0f * b0.y;
    Bb1.x = hiHalf ? 1.0f : -2.0f * b1.x;
    Bb1.y = hiHalf ? (b1.x * b1.x + b1.y * b1.y) : -2.0f * b1.y;
    Bd0.x = hiHalf ? 1.0f : -2.0f * d0.x;
    Bd0.y = hiHalf ? (d0.x * d0.x + d0.y * d0.y) : -2.0f * d0.y;
    Bd1.x = hiHalf ? 1.0f : -2.0f * d1.x;
    Bd1.y = hiHalf ? (d1.x * d1.x + d1.y * d1.y) : -2.0f * d1.y;

    v8f z = {0.f, 0.f, 0.f, 0.f, 0.f, 0.f, 0.f, 0.f};
    // squared-distance tiles: D[grid][corner]
    v8f Db0 = __builtin_amdgcn_wmma_f32_16x16x4_f32(false, A, false, Bb0,
                                                    (short)0, z, false, false);
    v8f Db1 = __builtin_amdgcn_wmma_f32_16x16x4_f32(false, A, false, Bb1,
                                                    (short)0, z, false, false);
    v8f Dd0 = __builtin_amdgcn_wmma_f32_16x16x4_f32(false, A, false, Bd0,
                                                    (short)0, z, false, false);
    v8f Dd1 = __builtin_amdgcn_wmma_f32_16x16x4_f32(false, A, false, Bd1,
                                                    (short)0, z, false, false);

    // min over corners n, n+16 and all 16 grid rows: 8 VGPRs in-lane, then cross-half (M 0..7 vs 8..15)
    float mb = xhalf_min(pairmin8(Db0, Db1));
    float md = xhalf_min(pairmin8(Dd0, Dd1));

    const float d2 = fmaxf(mb, md);          // max(min_b, min_d) of squared dist
    acc = fmaf(mt[4 + i], __expf(-200.0f * d2), acc);  // 0.5/bw^2 = 200
  }

  if (!hiHalf && g < GSIZE) out[(size_t)s * GSIZE + g] = acc;
}

extern "C" void kernel_launch(void* const* d_in, const int* in_sizes, int n_in,
                              void* d_out, int out_size, void* d_ws, size_t ws_size,
                              hipStream_t stream) {
  const float* births = (const float*)d_in[0];
  const float* deaths = (const float*)d_in[1];
  float* out  = (float*)d_out;
  float* meta = (float*)d_ws;

  const int S = in_sizes[0] / (NI * NC * 2);   // 128 samples

  pi_setup_kernel<<<dim3(S), dim3(256), 0, stream>>>(births, deaths, meta);

  const int tiles = (GSIZE + 15) / 16;         // 57 tiles of 16 grid points
  pi_main_kernel<<<dim3(tiles, S), dim3(32), 0, stream>>>(births, deaths, meta, out);
}